// QueryExecutor_88854283420258
// MI455X (gfx1250) — compile-verified
//
#include <hip/hip_runtime.h>
#include <cstdint>

#define PROJ_BIT (1 << 26)
#define BATCH 32

// ---------------------------------------------------------------------------
// Kernel 1: zero both score planes (S1 lives in d_out, S2 in d_ws).
// ---------------------------------------------------------------------------
__global__ __launch_bounds__(256) void qe_zero(float* __restrict__ s1,
                                               float* __restrict__ s2, int n4) {
  int i = blockIdx.x * blockDim.x + threadIdx.x;
  if (i < n4) {
    float4 z = make_float4(0.f, 0.f, 0.f, 0.f);
    reinterpret_cast<float4*>(s1)[i] = z;
    reinterpret_cast<float4*>(s2)[i] = z;
  }
}

// ---------------------------------------------------------------------------
// Kernel 2: sparse relational scatter.
// Query table staged to LDS via the gfx1250 async-LDS DMA path
// (global_load_async_to_lds_b32 + s_wait_asynccnt), then each thread scans
// edges and tests the packed (rel,head) key against all 32 batch rows.
// Matches are ~0.25 per (head,rel) pair, so atomics are essentially free and
// edge_t / edge_weight are only fetched on a hit.
// ---------------------------------------------------------------------------
__global__ __launch_bounds__(256) void qe_scatter(
    const int* __restrict__ query, const int* __restrict__ eh,
    const int* __restrict__ et, const int* __restrict__ er,
    const float* __restrict__ ew, float* __restrict__ s1,
    float* __restrict__ s2, int n_edge, int n_node) {
  __shared__ int sq[BATCH * 6];
  __shared__ int skey1[BATCH];
  __shared__ int skey2[BATCH];

  const int t = threadIdx.x;

  // Async DMA of the [32,6] query token table into LDS (threads 0..191,
  // i.e. waves 0..5 fully active -> uniform EXEC per issuing wave).
  if (t < BATCH * 6) {
    uint32_t lds_off = (uint32_t)(uintptr_t)(&sq[t]);  // low 32b = LDS offset
    const int* gaddr = query + t;
    asm volatile("global_load_async_to_lds_b32 %0, %1, off"
                 :: "v"(lds_off), "v"(gaddr)
                 : "memory");
  }
  asm volatile("s_wait_asynccnt 0" ::: "memory");
  __syncthreads();

  // Decode tokens: row = [e1, -(PROJ+r1), e2, -(PROJ+r2), -INTER, STOP].
  if (t < BATCH) {
    int e1 = sq[t * 6 + 0];
    int r1 = (-sq[t * 6 + 1]) & ~PROJ_BIT;
    int e2 = sq[t * 6 + 2];
    int r2 = (-sq[t * 6 + 3]) & ~PROJ_BIT;
    skey1[t] = (r1 << 17) | e1;  // n_node = 40000 < 2^17, n_rel = 64
    skey2[t] = (r2 << 17) | e2;
  }
  __syncthreads();

  for (int e = blockIdx.x * blockDim.x + t; e < n_edge;
       e += gridDim.x * blockDim.x) {
    int key = (er[e] << 17) | eh[e];
    unsigned m1 = 0u, m2 = 0u;
#pragma unroll
    for (int b = 0; b < BATCH; ++b) {
      m1 |= (key == skey1[b]) ? (1u << b) : 0u;
      m2 |= (key == skey2[b]) ? (1u << b) : 0u;
    }
    if (m1 | m2) {  // rare: expected 0.25 matched batches per (h,r)
      int tl = et[e];
      float w = ew[e];
      while (m1) {
        int b = __ffs(m1) - 1;
        m1 &= m1 - 1u;
        unsafeAtomicAdd(&s1[(size_t)b * n_node + tl], w);
      }
      while (m2) {
        int b = __ffs(m2) - 1;
        m2 &= m2 - 1u;
        unsafeAtomicAdd(&s2[(size_t)b * n_node + tl], w);
      }
    }
  }
}

// ---------------------------------------------------------------------------
// Kernel 3: out = sigmoid(S1) * sigmoid(S2), in place over d_out (S1).
// ---------------------------------------------------------------------------
__device__ __forceinline__ float qe_sig(float x) {
  return 1.0f / (1.0f + __expf(-x));
}

__global__ __launch_bounds__(256) void qe_final(float* __restrict__ out,
                                                const float* __restrict__ s2,
                                                int n4) {
  int i = blockIdx.x * blockDim.x + threadIdx.x;
  if (i >= n4) return;
  float4 a = reinterpret_cast<const float4*>(out)[i];
  float4 b = reinterpret_cast<const float4*>(s2)[i];
  float4 r;
  r.x = qe_sig(a.x) * qe_sig(b.x);
  r.y = qe_sig(a.y) * qe_sig(b.y);
  r.z = qe_sig(a.z) * qe_sig(b.z);
  r.w = qe_sig(a.w) * qe_sig(b.w);
  reinterpret_cast<float4*>(out)[i] = r;
}

// ---------------------------------------------------------------------------
// Inputs (setup_inputs order): query[32,6] i32, edge_h[E] i32, edge_t[E] i32,
// edge_r[E] i32, edge_weight[E] f32, num_node (scalar i32).
// Output: [32, N] f32.
// ---------------------------------------------------------------------------
extern "C" void kernel_launch(void* const* d_in, const int* in_sizes, int n_in,
                              void* d_out, int out_size, void* d_ws,
                              size_t ws_size, hipStream_t stream) {
  const int* query = (const int*)d_in[0];
  const int* eh = (const int*)d_in[1];
  const int* et = (const int*)d_in[2];
  const int* er = (const int*)d_in[3];
  const float* ew = (const float*)d_in[4];

  const int n_edge = in_sizes[1];
  const int n_node = out_size / BATCH;

  float* s1 = (float*)d_out;  // reused as accumulator, finalized in place
  float* s2 = (float*)d_ws;   // needs BATCH*n_node*4 = 5.12 MB of scratch

  const int n4 = out_size / 4;  // 1,280,000 / 4
  const int zb = (n4 + 255) / 256;
  const int sb = (n_edge + 255) / 256;

  qe_zero<<<zb, 256, 0, stream>>>(s1, s2, n4);
  qe_scatter<<<sb, 256, 0, stream>>>(query, eh, et, er, ew, s1, s2, n_edge,
                                     n_node);
  qe_final<<<zb, 256, 0, stream>>>(s1, s2, n4);
}